// Model_39676907881857
// MI455X (gfx1250) — compile-verified
//
#include <hip/hip_runtime.h>

// ---------------------------------------------------------------------------
// MI455X (gfx1250, wave32) implementation of the reference DAG.
//
//   t1 = input@w1   t2 = input@w2   t3 = w3@w3        (plain WMMA GEMMs)
//   P  = (t1*w1)@input   Q = t1@input                 (dual-acc WMMA GEMM,
//                                                      epilogue fuses the
//                                                      whole elementwise tail
//                                                      and writes t15)
//
// V_WMMA_F32_16X16X4_F32 throughout (fp32 matrix pipe = reference precision).
// Double-buffered LDS (1 barrier/stage), k-pair-interleaved B layout so every
// fragment load is a single aligned ds_load_b64 (no VGPR repack movs).
// ---------------------------------------------------------------------------

typedef __attribute__((ext_vector_type(2))) float v2f;
typedef __attribute__((ext_vector_type(8))) float v8f;

#define NN 2048
constexpr int BM  = 128;          // block tile M
constexpr int BN  = 128;          // block tile N
constexpr int BK  = 16;           // K stage depth (4 wmma k-steps)
constexpr int BPR = BK / 2;       // B pair-rows per stage
constexpr int BPS = 2 * BN + 32;  // padded pair-row stride (floats): half-waves
                                  // of a b64 frag load hit disjoint bank halves

// ---- per-wave geometry ----------------------------------------------------
struct WaveGeom {
  int lrow, half;           // lane%16, lane/16
  int wRow, wCol;           // wave tile origin inside block tile
};
__device__ __forceinline__ WaveGeom wave_geom(int tid) {
  WaveGeom g;
  const int lane = tid & 31;
  g.lrow = lane & 15;
  g.half = lane >> 4;
  const int wave = tid >> 5;        // 8 waves: 4 (M) x 2 (N)
  g.wRow = (wave >> 1) * 32;        // 32 rows per wave
  g.wCol = (wave & 1) * 64;         // 64 cols per wave
  return g;
}

// ---- plain C = A @ B (2048x2048, fp32 WMMA) -------------------------------
__global__ __launch_bounds__(256)
void wmma_gemm2048(float* __restrict__ C, const float* __restrict__ A,
                   const float* __restrict__ B) {
  __shared__ float As[2][BM * BK];      // 2 x 8 KB
  __shared__ float Bs[2][BPR * BPS];    // 2 x 9 KB, k-pair interleaved

  const int tid = threadIdx.x;
  const WaveGeom g = wave_geom(tid);
  const int blockRow = blockIdx.y * BM;
  const int blockCol = blockIdx.x * BN;

  // B staging geometry: thread owns k-rows {2*pr, 2*pr+1}, cols co..co+3
  const int pr = tid >> 5;
  const int co = (tid & 31) * 4;

  auto stage = [&](int buf, int k0) {
#pragma unroll
    for (int s = 0; s < 2; ++s) {       // A tile 128x16: 2 float4 per thread
      const int slot = tid * 2 + s;
      const int row = slot >> 2, c4 = (slot & 3) * 4;
      float4 v = *(const float4*)(A + (size_t)(blockRow + row) * NN + k0 + c4);
      *(float4*)(&As[buf][row * BK + c4]) = v;
    }
    {                                    // B tile 16x128, pair-interleaved
      const float* b0 = B + (size_t)(k0 + 2 * pr) * NN + blockCol + co;
      float4 r0 = *(const float4*)(b0);
      float4 r1 = *(const float4*)(b0 + NN);
      float* dst = &Bs[buf][pr * BPS + co * 2];
      *(float4*)(dst)     = make_float4(r0.x, r1.x, r0.y, r1.y);
      *(float4*)(dst + 4) = make_float4(r0.z, r1.z, r0.w, r1.w);
    }
    if (k0 + BK < NN) {  // gfx1250 global_prefetch_b8 hint for stage after next
      __builtin_prefetch(A + (size_t)(blockRow + (tid >> 1)) * NN + (k0 + BK), 0, 1);
      __builtin_prefetch(B + (size_t)(k0 + BK + 2 * pr) * NN + blockCol + co, 0, 1);
    }
  };

  auto compute = [&](int buf, v8f acc[2][4]) {
#pragma unroll
    for (int ks = 0; ks < 4; ++ks) {
      v2f af[2], bf[4];
#pragma unroll
      for (int m = 0; m < 2; ++m)
        af[m] = *(const v2f*)(&As[buf][(g.wRow + m * 16 + g.lrow) * BK +
                                       ks * 4 + g.half * 2]);
#pragma unroll
      for (int n = 0; n < 4; ++n)
        bf[n] = *(const v2f*)(&Bs[buf][(ks * 2 + g.half) * BPS +
                                       (g.wCol + n * 16 + g.lrow) * 2]);
#pragma unroll
      for (int m = 0; m < 2; ++m)
#pragma unroll
        for (int n = 0; n < 4; ++n)
          acc[m][n] = __builtin_amdgcn_wmma_f32_16x16x4_f32(
              false, af[m], false, bf[n], (short)0, acc[m][n], false, false);
    }
  };

  v8f acc[2][4] = {};
  stage(0, 0);
  int cur = 0;
  for (int k0 = 0; k0 < NN; k0 += BK) {
    __syncthreads();                        // buf[cur] staged, old reads done
    if (k0 + BK < NN) stage(cur ^ 1, k0 + BK);
    compute(cur, acc);
    cur ^= 1;
  }

#pragma unroll
  for (int m = 0; m < 2; ++m)
#pragma unroll
    for (int n = 0; n < 4; ++n)
#pragma unroll
      for (int j = 0; j < 8; ++j) {
        const int r = blockRow + g.wRow + m * 16 + g.half * 8 + j;
        const int c = blockCol + g.wCol + n * 16 + g.lrow;
        C[(size_t)r * NN + c] = acc[m][n][j];
      }
}

// ---- elementwise tail of the DAG ------------------------------------------
// a=t1, b=t2, c=t3, p=(t1*w1)@input, q=t1@input, w=w1
__device__ __forceinline__ float dag_combine(float a, float b, float c,
                                             float p, float q, float w) {
  float t6  = p * w;
  float t7  = a * b + a + c;
  float t8  = b - b;              // t2 - t4 with t4 == t2 bit-exact
  float t9  = t7 * t8 + t6 * c;
  float t11 = c * t9;
  float t12 = c * q * t6;
  return (t7 + t11) + (t8 + t12); // t15
}

// ---- dual GEMM (P and Q share A/B tiles) + fused final epilogue -----------
__global__ __launch_bounds__(256)
void wmma_dual_gemm_final(const float* __restrict__ T1, const float* __restrict__ W1,
                          const float* __restrict__ In, const float* __restrict__ T2,
                          const float* __restrict__ T3, float* __restrict__ Out) {
  __shared__ float AsQ[2][BM * BK];     // t1 tile
  __shared__ float AsP[2][BM * BK];     // (t1*w1) tile
  __shared__ float Bs[2][BPR * BPS];    // input tile, k-pair interleaved

  const int tid = threadIdx.x;
  const WaveGeom g = wave_geom(tid);
  const int blockRow = blockIdx.y * BM;
  const int blockCol = blockIdx.x * BN;

  const int pr = tid >> 5;
  const int co = (tid & 31) * 4;

  auto stage = [&](int buf, int k0) {
#pragma unroll
    for (int s = 0; s < 2; ++s) {
      const int slot = tid * 2 + s;
      const int row = slot >> 2, c4 = (slot & 3) * 4;
      const size_t off = (size_t)(blockRow + row) * NN + k0 + c4;
      float4 a = *(const float4*)(T1 + off);
      float4 w = *(const float4*)(W1 + off);
      *(float4*)(&AsQ[buf][row * BK + c4]) = a;
      *(float4*)(&AsP[buf][row * BK + c4]) =
          make_float4(a.x * w.x, a.y * w.y, a.z * w.z, a.w * w.w);
    }
    {
      const float* b0 = In + (size_t)(k0 + 2 * pr) * NN + blockCol + co;
      float4 r0 = *(const float4*)(b0);
      float4 r1 = *(const float4*)(b0 + NN);
      float* dst = &Bs[buf][pr * BPS + co * 2];
      *(float4*)(dst)     = make_float4(r0.x, r1.x, r0.y, r1.y);
      *(float4*)(dst + 4) = make_float4(r0.z, r1.z, r0.w, r1.w);
    }
    if (k0 + BK < NN) {
      __builtin_prefetch(T1 + (size_t)(blockRow + (tid >> 1)) * NN + (k0 + BK), 0, 1);
      __builtin_prefetch(W1 + (size_t)(blockRow + (tid >> 1)) * NN + (k0 + BK), 0, 1);
      __builtin_prefetch(In + (size_t)(k0 + BK + 2 * pr) * NN + blockCol + co, 0, 1);
    }
  };

  v8f accP[2][4] = {};
  v8f accQ[2][4] = {};

  stage(0, 0);
  int cur = 0;
  for (int k0 = 0; k0 < NN; k0 += BK) {
    __syncthreads();
    if (k0 + BK < NN) stage(cur ^ 1, k0 + BK);
#pragma unroll
    for (int ks = 0; ks < 4; ++ks) {
      v2f afQ[2], afP[2], bf[4];
#pragma unroll
      for (int m = 0; m < 2; ++m) {
        const int ao = (g.wRow + m * 16 + g.lrow) * BK + ks * 4 + g.half * 2;
        afQ[m] = *(const v2f*)(&AsQ[cur][ao]);
        afP[m] = *(const v2f*)(&AsP[cur][ao]);
      }
#pragma unroll
      for (int n = 0; n < 4; ++n)
        bf[n] = *(const v2f*)(&Bs[cur][(ks * 2 + g.half) * BPS +
                                       (g.wCol + n * 16 + g.lrow) * 2]);
#pragma unroll
      for (int m = 0; m < 2; ++m)
#pragma unroll
        for (int n = 0; n < 4; ++n) {
          accP[m][n] = __builtin_amdgcn_wmma_f32_16x16x4_f32(
              false, afP[m], false, bf[n], (short)0, accP[m][n], false, false);
          accQ[m][n] = __builtin_amdgcn_wmma_f32_16x16x4_f32(
              false, afQ[m], false, bf[n], (short)0, accQ[m][n], false, false);
        }
    }
    cur ^= 1;
  }

  // Fused elementwise epilogue: write t15 directly.
#pragma unroll
  for (int m = 0; m < 2; ++m)
#pragma unroll
    for (int n = 0; n < 4; ++n)
#pragma unroll
      for (int j = 0; j < 8; ++j) {
        const int r = blockRow + g.wRow + m * 16 + g.half * 8 + j;
        const int c = blockCol + g.wCol + n * 16 + g.lrow;
        const size_t off = (size_t)r * NN + c;
        const float a  = T1[off];
        const float b  = T2[off];
        const float c3 = T3[off];
        const float w  = W1[off];
        Out[off] = dag_combine(a, b, c3, accP[m][n][j], accQ[m][n][j], w);
      }
}

extern "C" void kernel_launch(void* const* d_in, const int* in_sizes, int n_in,
                              void* d_out, int out_size, void* d_ws, size_t ws_size,
                              hipStream_t stream) {
  (void)in_sizes; (void)n_in; (void)out_size; (void)ws_size;
  const float* input = (const float*)d_in[0];
  const float* w1    = (const float*)d_in[1];
  const float* w2    = (const float*)d_in[2];
  const float* w3    = (const float*)d_in[3];
  float* out = (float*)d_out;
  float* ws  = (float*)d_ws;

  const size_t M = (size_t)NN * NN;   // 4M elements = 16 MB per matrix
  float* t1 = ws;                     // input @ w1
  float* t2 = ws + M;                 // input @ w2
  float* t3 = ws + 2 * M;             // w3 @ w3        (48 MB workspace total)

  dim3 grid(NN / BN, NN / BM);
  dim3 block(256);                    // 8 waves (wave32)

  wmma_gemm2048<<<grid, block, 0, stream>>>(t1, input, w1);
  wmma_gemm2048<<<grid, block, 0, stream>>>(t2, input, w2);
  wmma_gemm2048<<<grid, block, 0, stream>>>(t3, w3, w3);
  wmma_dual_gemm_final<<<grid, block, 0, stream>>>(t1, w1, input, t2, t3, out);
}